// TransformerLayer_16054587752894
// MI455X (gfx1250) — compile-verified
//
#include <hip/hip_runtime.h>

// ---------------------------------------------------------------------------
// MI455X (gfx1250) implementation.
//  - f32 WMMA (v_wmma_f32_16x16x4_f32) for all GEMMs (exact f32 math).
//  - hardware v_tanh_f32 / v_exp_f32 for the Bahdanau score tensor.
//  - LDS staging of k-rows (65-float pitch => conflict-free) for the
//    O(N*L*L*D) score stage, which dominates the runtime.
// ---------------------------------------------------------------------------

typedef float v2f __attribute__((ext_vector_type(2)));
typedef float v8f __attribute__((ext_vector_type(8)));

constexpr int N_ = 8, C_ = 128, L_ = 512, D_ = 64, F_ = 512;
constexpr int BW = 64;       // attention band width
constexpr int HB = BW / 2;   // 32
constexpr float LN_EPS_ = 1e-5f;
constexpr float ATTN_EPS_ = 1e-6f;

#define DEVINL __device__ __forceinline__

DEVINL float tanh_hw(float x) {
  float r;
  // CDNA5 hardware tanh (trans op). v_nop covers the trans result-use
  // hazard (ISA 7.4: 1 independent op after a trans op before use).
  asm("v_tanh_f32 %0, %1\n\tv_nop" : "=v"(r) : "v"(x));
  return r;
}

DEVINL v8f wmma_f32_k4(v2f a, v2f b, v8f c) {
  // D = A(16x4, f32) * B(4x16, f32) + C(16x16, f32)
  return __builtin_amdgcn_wmma_f32_16x16x4_f32(false, a, false, b, (short)0, c,
                                               false, false);
}

// ---------------------------------------------------------------------------
// Kernel 1: q = xt @ Wt, k = xt @ Wx.  One 16x16 output tile per wave.
// x is (N,C,L): element (M=l, K=c) is contiguous along lanes (l). Output
// stored (N,L,D) so kernel 2 gets coalesced 64-float rows.
// ---------------------------------------------------------------------------
__global__ __launch_bounds__(256) void qk_proj_kernel(
    const float* __restrict__ x, const float* __restrict__ Wt,
    const float* __restrict__ Wx, float* __restrict__ q,
    float* __restrict__ k) {
  int wid = (blockIdx.x * 256 + (int)threadIdx.x) >> 5;  // 0..2047
  int lane = threadIdx.x & 31, hv = lane >> 4, l16 = lane & 15;
  int m = wid >> 3;                // row tile 0..255
  int rem = wid & 7;
  int d0 = (rem >> 1) << 4;        // 0,16,32,48
  int which = rem & 1;             // 0: q (Wt), 1: k (Wx)
  const float* W = which ? Wx : Wt;
  float* out = which ? k : q;
  int n = m >> 5, l0 = (m & 31) << 4;
  const float* xb = x + (size_t)n * C_ * L_;
  v8f acc = {};
  for (int k0 = 0; k0 < C_; k0 += 4) {
    int c = k0 + 2 * hv;  // A: lanes16-31 hold K+2/K+3 ; B mirrored
    v2f a, b;
    a.x = xb[(size_t)c * L_ + l0 + l16];
    a.y = xb[(size_t)(c + 1) * L_ + l0 + l16];
    b.x = W[(size_t)c * D_ + d0 + l16];
    b.y = W[(size_t)(c + 1) * D_ + d0 + l16];
    acc = wmma_f32_k4(a, b, acc);
  }
#pragma unroll
  for (int r = 0; r < 8; ++r) {
    int l = l0 + r + 8 * hv;  // D tile: lanes16-31 hold rows M+8
    out[((size_t)(n * L_ + l)) * D_ + d0 + l16] = acc[r];
  }
}

// ---------------------------------------------------------------------------
// Kernel 2: scores, softmax (global row max, banded sum), attention output
// 'a', banded v = a@xt, residual + LayerNorm0.  Block = 256 thr = 16 rows x
// 16 column-threads; each thread owns 32 j-columns (j = ch*128 + tcol + 16*jj).
// ---------------------------------------------------------------------------
__global__ __launch_bounds__(256) void attn_kernel(
    const float* __restrict__ x, const float* __restrict__ q,
    const float* __restrict__ k, const float* __restrict__ bh,
    const float* __restrict__ Wa, const float* __restrict__ g0,
    const float* __restrict__ be0, float* __restrict__ aout,
    float* __restrict__ x2s) {
  __shared__ float ksh[128][65];   // 65-pitch: lane j-stride 1 -> no conflicts
  __shared__ float qsh[16][64];
  __shared__ float wash[64];
  __shared__ float band[16][64];   // unnormalized band weights per row
  __shared__ float red[16][17];
  __shared__ float redq[16][17];

  int n = blockIdx.y, i0 = blockIdx.x * 16;
  int tid = threadIdx.x;
  for (int idx = tid; idx < 16 * 64; idx += 256) {
    int r = idx >> 6, d = idx & 63;
    qsh[r][d] = q[((size_t)(n * L_ + i0 + r)) * D_ + d] + bh[d];
    band[r][d] = 0.f;
  }
  if (tid < 64) wash[tid] = Wa[tid];

  int row = tid >> 4, tcol = tid & 15;
  int i = i0 + row;
  float acc[32];

#pragma unroll
  for (int ch = 0; ch < 4; ++ch) {
    __syncthreads();
    for (int idx = tid; idx < 128 * 64; idx += 256) {
      int jr = idx >> 6, d = idx & 63;
      ksh[jr][d] = k[((size_t)(n * L_ + ch * 128 + jr)) * D_ + d];
    }
    __syncthreads();
    float e8[8] = {};
#pragma unroll 4
    for (int d = 0; d < 64; ++d) {
      float qv = qsh[row][d];
      float wv = wash[d];
#pragma unroll
      for (int jj = 0; jj < 8; ++jj)
        e8[jj] += wv * tanh_hw(qv + ksh[tcol + 16 * jj][d]);
    }
#pragma unroll
    for (int jj = 0; jj < 8; ++jj) acc[ch * 8 + jj] = e8[jj];
  }

  // ---- global row max over all 512 columns (reference takes max pre-mask)
  float mloc = -3.4e38f;
#pragma unroll
  for (int kk = 0; kk < 32; ++kk) mloc = fmaxf(mloc, acc[kk]);
  red[row][tcol] = mloc;
  __syncthreads();
  if (tcol == 0) {
    float mm = red[row][0];
    for (int t = 1; t < 16; ++t) mm = fmaxf(mm, red[row][t]);
    red[row][16] = mm;
  }
  __syncthreads();
  float M = red[row][16];

  // ---- banded exp + sum (ba cancels exactly in exp(e-max))
  int jlo = i - HB;
  float sloc = 0.f;
#pragma unroll
  for (int ch = 0; ch < 4; ++ch)
#pragma unroll
    for (int jj = 0; jj < 8; ++jj) {
      int j = ch * 128 + tcol + 16 * jj;
      int off = j - jlo;
      float w = 0.f;
      if (off >= 0 && off < BW) {
        w = __builtin_amdgcn_exp2f((acc[ch * 8 + jj] - M) * 1.4426950408889634f);
        band[row][off] = w;
        sloc += w;
      }
      acc[ch * 8 + jj] = w;
    }
  red[row][tcol] = sloc;
  __syncthreads();
  if (tcol == 0) {
    float s = 0.f;
    for (int t = 0; t < 16; ++t) s += red[row][t];
    red[row][16] = s;
  }
  __syncthreads();
  float inv = 1.0f / (red[row][16] + ATTN_EPS_);

  // ---- write full attention row (zeros outside band; d_out is poisoned)
  float* arow = aout + ((size_t)(n * L_ + i)) * L_;
#pragma unroll
  for (int ch = 0; ch < 4; ++ch)
#pragma unroll
    for (int jj = 0; jj < 8; ++jj) {
      int j = ch * 128 + tcol + 16 * jj;
      arow[j] = acc[ch * 8 + jj] * inv;
    }
  __syncthreads();

  // ---- v = a @ xt over band, residual, LayerNorm0; store x2 as (N,C,L)
  const float* xn = x + (size_t)n * C_ * L_;
  float sv[8] = {};
  for (int jj = 0; jj < BW; ++jj) {
    float bw = band[row][jj];
    int j = jlo + jj;
    int jc = j < 0 ? 0 : (j > L_ - 1 ? L_ - 1 : j);  // bw==0 when clipped
#pragma unroll
    for (int kk = 0; kk < 8; ++kk)
      sv[kk] += bw * xn[(size_t)(tcol + 16 * kk) * L_ + jc];
  }
  float ps = 0.f, pq = 0.f;
#pragma unroll
  for (int kk = 0; kk < 8; ++kk) {
    int c = tcol + 16 * kk;
    float val = sv[kk] * inv + xn[(size_t)c * L_ + i];
    sv[kk] = val;
    ps += val;
    pq += val * val;
  }
  red[row][tcol] = ps;
  redq[row][tcol] = pq;
  __syncthreads();
  if (tcol == 0) {
    float s = 0.f, sq = 0.f;
    for (int t = 0; t < 16; ++t) { s += red[row][t]; sq += redq[row][t]; }
    float mu = s * (1.0f / C_);
    float var = sq * (1.0f / C_) - mu * mu;
    red[row][16] = mu;
    redq[row][16] = 1.0f / sqrtf(var + LN_EPS_);
  }
  __syncthreads();
  float mu = red[row][16], rs = redq[row][16];
#pragma unroll
  for (int kk = 0; kk < 8; ++kk) {
    int c = tcol + 16 * kk;
    x2s[((size_t)(n * C_ + c)) * L_ + i] = (sv[kk] - mu) * rs * g0[c] + be0[c];
  }
}

// ---------------------------------------------------------------------------
// Kernel 3: t = relu(x2 @ W0 + b0), WMMA, one 16x16 tile per wave.
// t stored f-major (N,F,L) so kernel 4's A loads are lane-contiguous.
// ---------------------------------------------------------------------------
__global__ __launch_bounds__(256) void ffn1_kernel(
    const float* __restrict__ x2s, const float* __restrict__ W0,
    const float* __restrict__ b0, float* __restrict__ ts) {
  int wid = (blockIdx.x * 256 + (int)threadIdx.x) >> 5;  // 0..8191
  int lane = threadIdx.x & 31, hv = lane >> 4, l16 = lane & 15;
  int m = wid >> 5;           // row tile 0..255
  int f0 = (wid & 31) << 4;   // 0..496
  int n = m >> 5, l0 = (m & 31) << 4;
  const float* xb = x2s + (size_t)n * C_ * L_;
  v8f acc = {};
  for (int k0 = 0; k0 < C_; k0 += 4) {
    int c = k0 + 2 * hv;
    v2f a, b;
    a.x = xb[(size_t)c * L_ + l0 + l16];
    a.y = xb[(size_t)(c + 1) * L_ + l0 + l16];
    b.x = W0[(size_t)c * F_ + f0 + l16];
    b.y = W0[(size_t)(c + 1) * F_ + f0 + l16];
    acc = wmma_f32_k4(a, b, acc);
  }
  float bv = b0[f0 + l16];
#pragma unroll
  for (int r = 0; r < 8; ++r) {
    int l = l0 + r + 8 * hv;
    ts[((size_t)(n * F_ + f0 + l16)) * L_ + l] = fmaxf(acc[r] + bv, 0.f);
  }
}

// ---------------------------------------------------------------------------
// Kernel 4: x3 = t @ W1 + b1 ; x4 = LN(x3 + x2) ; store transposed (N,C,L).
// Block = 8 waves: wave w computes the 16x16 tile for columns c0 = 16w of the
// same 16 rows, so the whole 16x128 row block is in LDS for the LayerNorm.
// ---------------------------------------------------------------------------
__global__ __launch_bounds__(256) void ffn2_kernel(
    const float* __restrict__ ts, const float* __restrict__ W1,
    const float* __restrict__ b1, const float* __restrict__ x2s,
    const float* __restrict__ g1, const float* __restrict__ be1,
    float* __restrict__ out0) {
  __shared__ float sh[16][132];
  __shared__ float red[16][17];
  __shared__ float redq[16][17];
  int m = blockIdx.x;  // 0..255
  int n = m >> 5, l0 = (m & 31) << 4;
  int tid = threadIdx.x;
  int lane = tid & 31, hv = lane >> 4, l16 = lane & 15;
  int c0 = (tid >> 5) << 4;
  const float* tb = ts + (size_t)n * F_ * L_;
  v8f acc = {};
  for (int k0 = 0; k0 < F_; k0 += 4) {
    int f = k0 + 2 * hv;
    v2f a, b;
    a.x = tb[(size_t)f * L_ + l0 + l16];
    a.y = tb[(size_t)(f + 1) * L_ + l0 + l16];
    b.x = W1[(size_t)f * C_ + c0 + l16];
    b.y = W1[(size_t)(f + 1) * C_ + c0 + l16];
    acc = wmma_f32_k4(a, b, acc);
  }
  float bv = b1[c0 + l16];
#pragma unroll
  for (int r = 0; r < 8; ++r) sh[r + 8 * hv][c0 + l16] = acc[r] + bv;
  __syncthreads();
  // residual += x2 (coalesced reads from (N,C,L) layout)
  for (int idx = tid; idx < 16 * C_; idx += 256) {
    int c = idx >> 4, ll = idx & 15;
    sh[ll][c] += x2s[((size_t)(n * C_ + c)) * L_ + l0 + ll];
  }
  __syncthreads();
  // LayerNorm1 per row (16 threads per row)
  int row = tid >> 4, tcol = tid & 15;
  float ps = 0.f, pq = 0.f;
#pragma unroll
  for (int kk = 0; kk < 8; ++kk) {
    float v = sh[row][tcol + 16 * kk];
    ps += v;
    pq += v * v;
  }
  red[row][tcol] = ps;
  redq[row][tcol] = pq;
  __syncthreads();
  if (tcol == 0) {
    float s = 0.f, sq = 0.f;
    for (int t = 0; t < 16; ++t) { s += red[row][t]; sq += redq[row][t]; }
    float mu = s * (1.0f / C_);
    float var = sq * (1.0f / C_) - mu * mu;
    red[row][16] = mu;
    redq[row][16] = 1.0f / sqrtf(var + LN_EPS_);
  }
  __syncthreads();
  float mu = red[row][16], rs = redq[row][16];
#pragma unroll
  for (int kk = 0; kk < 8; ++kk) {
    int c = tcol + 16 * kk;
    sh[row][c] = (sh[row][c] - mu) * rs * g1[c] + be1[c];
  }
  __syncthreads();
  // transposed, coalesced store: out0[n][c][l]
  for (int idx = tid; idx < 16 * C_; idx += 256) {
    int c = idx >> 4, ll = idx & 15;
    out0[((size_t)(n * C_ + c)) * L_ + l0 + ll] = sh[ll][c];
  }
}

// ---------------------------------------------------------------------------
extern "C" void kernel_launch(void* const* d_in, const int* in_sizes, int n_in,
                              void* d_out, int out_size, void* d_ws,
                              size_t ws_size, hipStream_t stream) {
  (void)in_sizes; (void)n_in; (void)out_size; (void)ws_size;
  const float* x   = (const float*)d_in[0];
  const float* Wx  = (const float*)d_in[1];
  const float* Wt  = (const float*)d_in[2];
  const float* bh  = (const float*)d_in[3];
  const float* Wa  = (const float*)d_in[4];
  // d_in[5] = ba: cancels exactly in softmax (exp(e+ba-(M+ba)) == exp(e-M))
  const float* W0  = (const float*)d_in[6];
  const float* b0  = (const float*)d_in[7];
  const float* W1  = (const float*)d_in[8];
  const float* b1  = (const float*)d_in[9];
  const float* g0  = (const float*)d_in[10];
  const float* be0 = (const float*)d_in[11];
  const float* g1  = (const float*)d_in[12];
  const float* be1 = (const float*)d_in[13];

  float* out0 = (float*)d_out;                       // (N,C,L)
  float* aout = out0 + (size_t)N_ * C_ * L_;         // (N,L,L)

  float* qws = (float*)d_ws;                         // (N,L,D)
  float* kws = qws + (size_t)N_ * L_ * D_;           // (N,L,D)
  float* x2s = kws + (size_t)N_ * L_ * D_;           // (N,C,L)
  float* tsw = x2s + (size_t)N_ * C_ * L_;           // (N,F,L)

  qk_proj_kernel<<<256, 256, 0, stream>>>(x, Wt, Wx, qws, kws);
  attn_kernel<<<dim3(L_ / 16, N_), 256, 0, stream>>>(x, qws, kws, bh, Wa, g0,
                                                     be0, aout, x2s);
  ffn1_kernel<<<1024, 256, 0, stream>>>(x2s, W0, b0, tsw);
  ffn2_kernel<<<256, 256, 0, stream>>>(tsw, W1, b1, x2s, g1, be1, out0);
}